// SPDAvgPool_59227599012349
// MI455X (gfx1250) — compile-verified
//
#include <hip/hip_runtime.h>
#include <hip/hip_bf16.h>

// ---------------------------------------------------------------------------
// SPD matrix-log -> 2x2 avgpool -> matrix-exp, batched over 8192 64x64 SPD
// matrices.  Eigh-free: inverse scaling-and-squaring (Newton-Schulz sqrt) for
// logm, scaling-and-squaring Taylor for expm.  All matmuls via
// V_WMMA_F32_16X16X4_F32 on LDS-resident tiles (fp32 needed for stability).
// One workgroup (8 wave32) per matrix.
//
// Every operand of every matmul here is symmetric (all iterates are
// polynomials of a symmetric matrix), so B fragments are read ROW-major
// (B[n][k] == B[k][n]): both A and B fragments are consecutive-dword pairs,
// which lowers to paired ds_load_2addr into aligned VGPR pairs with no
// register shuffling before v_wmma.
// ---------------------------------------------------------------------------

typedef float v2f __attribute__((ext_vector_type(2)));
typedef float v8f __attribute__((ext_vector_type(8)));

#define S64 66      // LDS row stride (floats) for 64x64 buffers (bank-spread)
#define S32 34      // LDS row stride for 32x32 buffers
#define NS_ITERS 11 // Newton-Schulz iterations per square root
#define NSQRT 4     // number of repeated square roots (2^4 = 16)
#define EXP_TERMS 12

static __device__ __forceinline__ v8f wmma4(v2f a, v2f b, v8f c) {
  // D = A(16x4) * B(4x16) + C(16x16), fp32
  return __builtin_amdgcn_wmma_f32_16x16x4_f32(
      /*neg_a=*/false, a, /*neg_b=*/false, b,
      /*c_mod=*/(short)0, c, /*reuse_a=*/false, /*reuse_b=*/false);
}

// C(64x64) = alpha * A*B + beta*I, A,B symmetric, LDS stride S64.
// Wave w owns tile-row (w>>1), tile-cols {2*(w&1), 2*(w&1)+1}: A-frag shared.
// Trailing barrier only (every producer stage ends with a barrier).
static __device__ __forceinline__ void mm64(float* C, const float* A,
                                            const float* B, float alpha,
                                            float beta) {
  const int wave = threadIdx.x >> 5;
  const int lane = threadIdx.x & 31;
  const int n = lane & 15;     // M index for A-frag, N index for B/C-frags
  const int half = lane >> 4;  // selects K pairs / row-halves per ISA layout
  const int ti = wave >> 1;
  const int tj0 = (wave & 1) * 2;  // two adjacent tile columns
  v8f acc0 = {0.f, 0.f, 0.f, 0.f, 0.f, 0.f, 0.f, 0.f};
  v8f acc1 = acc0;
  const float* Arow = A + (ti * 16 + n) * S64;
  const float* Brow0 = B + (tj0 * 16 + n) * S64;        // symmetric read
  const float* Brow1 = B + (tj0 * 16 + 16 + n) * S64;   // symmetric read
  for (int k0 = 0; k0 < 64; k0 += 4) {
    const int ka = k0 + 2 * half;
    v2f af, bf0, bf1;
    af.x = Arow[ka];          // A: (m, ka+0)
    af.y = Arow[ka + 1];      // A: (m, ka+1)
    bf0.x = Brow0[ka];        // B: (ka+0, n) == B[n][ka]   tile tj0
    bf0.y = Brow0[ka + 1];
    bf1.x = Brow1[ka];        //                            tile tj0+1
    bf1.y = Brow1[ka + 1];
    acc0 = wmma4(af, bf0, acc0);
    acc1 = wmma4(af, bf1, acc1);
  }
#pragma unroll
  for (int v = 0; v < 8; ++v) {
    const int row = ti * 16 + v + 8 * half;
    const int col0 = tj0 * 16 + n;
    float v0 = alpha * acc0[v];
    float v1 = alpha * acc1[v];
    if (row == col0) v0 += beta;
    if (row == col0 + 16) v1 += beta;
    C[row * S64 + col0] = v0;
    C[row * S64 + col0 + 16] = v1;
  }
  __syncthreads();
}

// C(32x32) = alpha * A*B + beta*I, symmetric operands; waves 0..3 compute
static __device__ __forceinline__ void mm32(float* C, const float* A,
                                            const float* B, float alpha,
                                            float beta) {
  const int wave = threadIdx.x >> 5;
  const int lane = threadIdx.x & 31;
  const int n = lane & 15;
  const int half = lane >> 4;
  if (wave < 4) {  // wave-uniform branch: EXEC stays all-ones inside
    const int ti = wave >> 1;
    const int tj = wave & 1;
    v8f acc = {0.f, 0.f, 0.f, 0.f, 0.f, 0.f, 0.f, 0.f};
    const float* Arow = A + (ti * 16 + n) * S32;
    const float* Brow = B + (tj * 16 + n) * S32;  // symmetric read
    for (int k0 = 0; k0 < 32; k0 += 4) {
      const int ka = k0 + 2 * half;
      v2f af, bf;
      af.x = Arow[ka];
      af.y = Arow[ka + 1];
      bf.x = Brow[ka];
      bf.y = Brow[ka + 1];
      acc = wmma4(af, bf, acc);
    }
#pragma unroll
    for (int v = 0; v < 8; ++v) {
      const int row = ti * 16 + v + 8 * half;
      const int col = tj * 16 + n;
      float val = alpha * acc[v];
      if (row == col) val += beta;
      C[row * S32 + col] = val;
    }
  }
  __syncthreads();
}

// dst = a*src + bdiag*I over 64x64
static __device__ __forceinline__ void elemwise64(float* dst, const float* src,
                                                  float a, float bdiag) {
  for (int idx = threadIdx.x; idx < 4096; idx += 256) {
    const int r = idx >> 6, c = idx & 63;
    float v = a * src[r * S64 + c];
    if (r == c) v += bdiag;
    dst[r * S64 + c] = v;
  }
  __syncthreads();
}

// dst = bdiag*I over 64x64
static __device__ __forceinline__ void setIdent64(float* dst, float bdiag) {
  for (int idx = threadIdx.x; idx < 4096; idx += 256) {
    const int r = idx >> 6, c = idx & 63;
    dst[r * S64 + c] = (r == c) ? bdiag : 0.f;
  }
  __syncthreads();
}

__global__ __launch_bounds__(256) void spd_logpool_exp_kernel(
    const float* __restrict__ in, float* __restrict__ out) {
  __shared__ float buf[4 * 64 * S64];  // 4 x (64x66) fp32 work matrices
  __shared__ float red;

  const int b = blockIdx.x;
  const float* Ain = in + (size_t)b * 4096;

  float* P0 = buf;
  float* P1 = buf + 64 * S64;
  float* P2 = buf + 2 * 64 * S64;
  float* P3 = buf + 3 * 64 * S64;

  // ---- load A, compute Frobenius norm --------------------------------------
  if (threadIdx.x == 0) red = 0.f;
  __syncthreads();
  {
    float part = 0.f;
    for (int idx = threadIdx.x; idx < 4096; idx += 256) {
      const float v = Ain[idx];
      const int r = idx >> 6, c = idx & 63;
      P0[r * S64 + c] = v;
      part += v * v;
    }
    atomicAdd(&red, part);
  }
  __syncthreads();
  const float s = sqrtf(red);     // spectral norm upper bound; eig(A/s)<=1
  const float logs = logf(s);
  elemwise64(P0, P0, 1.0f / s, 0.f);  // Y = A/s

  // ---- inverse scaling-and-squaring: X = (A/s)^(1/2^NSQRT) -----------------
  float* Y = P0;
  float* Z = P1;
  float* T = P2;
  float* U = P3;
  for (int sq = 0; sq < NSQRT; ++sq) {
    setIdent64(Z, 1.f);  // Z = I
    for (int it = 0; it < NS_ITERS; ++it) {
      mm64(T, Z, Y, -0.5f, 1.5f);  // T = 1.5I - 0.5 Z*Y
      mm64(U, Y, T, 1.0f, 0.f);    // newY = Y*T
      mm64(Y, T, Z, 1.0f, 0.f);    // newZ = T*Z  (into old Y buffer)
      float* nY = U;
      float* nZ = Y;
      float* nU = Z;
      Y = nY; Z = nZ; U = nU;      // T unchanged; all four stay distinct
    }
  }

  // ---- log(X) via Mercator/Horner series, E = X - I ------------------------
  elemwise64(Z, Y, 1.f, -1.f);  // E in Z  (||E|| ~ 0.17)
  float* E = Z;
  elemwise64(T, E, -1.f / 6.f, 0.2f);  // P5 = c6*E + c5*I
  mm64(U, E, T, 1.f, -0.25f);          // P4
  mm64(T, E, U, 1.f, 1.f / 3.f);       // P3
  mm64(U, E, T, 1.f, -0.5f);           // P2
  mm64(T, E, U, 1.f, 1.f);             // P1
  mm64(U, E, T, 1.f, 0.f);             // L = E*P1   (logm(A/s)/2^NSQRT)
  float* L = U;

  // ---- fused 2x2 avgpool + 2^m scale + log(s)*I correction -----------------
  float* G = Y;   // 32x32 pooled log-matrix
  float* R = Z;   // exp ping buffer
  float* Tm = T;  // exp pong buffer
  if (threadIdx.x == 0) red = 0.f;
  __syncthreads();
  {
    const float scale2m = (float)(1 << NSQRT);
    float part = 0.f;
    for (int idx = threadIdx.x; idx < 1024; idx += 256) {
      const int r = idx >> 5, c = idx & 31;
      const float sum = L[(2 * r) * S64 + 2 * c] + L[(2 * r) * S64 + 2 * c + 1] +
                        L[(2 * r + 1) * S64 + 2 * c] +
                        L[(2 * r + 1) * S64 + 2 * c + 1];
      float val = 0.25f * scale2m * sum;
      if (r == c) val += 0.5f * logs;  // pooled identity contribution
      G[r * S32 + c] = val;
      part += val * val;
    }
    atomicAdd(&red, part);
  }
  __syncthreads();

  // ---- expm(G): scaling-and-squaring + Taylor-Horner -----------------------
  const float gn = sqrtf(red);  // ||G||_F >= ||G||_2
  int j = 0;
  {
    float t = gn;
    while (t > 0.5f && j < 24) { t *= 0.5f; ++j; }  // block-uniform
  }
  const float sc = exp2f(-(float)j);
  for (int idx = threadIdx.x; idx < 1024; idx += 256) {
    const int r = idx >> 5, c = idx & 31;
    G[r * S32 + c] *= sc;
  }
  // R = I (different buffer than G: no hazard with the scaling loop above)
  for (int idx = threadIdx.x; idx < 1024; idx += 256) {
    const int r = idx >> 5, c = idx & 31;
    R[r * S32 + c] = (r == c) ? 1.f : 0.f;
  }
  __syncthreads();
  for (int k = EXP_TERMS; k >= 1; --k) {  // R = I + (G*R)/k
    mm32(Tm, G, R, 1.0f / (float)k, 1.0f);
    float* tmp = R; R = Tm; Tm = tmp;
  }
  for (int q = 0; q < j; ++q) {  // undo scaling: R = R*R, j times
    mm32(Tm, R, R, 1.0f, 0.0f);
    float* tmp = R; R = Tm; Tm = tmp;
  }

  // ---- store 32x32 result --------------------------------------------------
  float* O = out + (size_t)b * 1024;
  for (int idx = threadIdx.x; idx < 1024; idx += 256) {
    const int r = idx >> 5, c = idx & 31;
    O[idx] = R[r * S32 + c];
  }
}

extern "C" void kernel_launch(void* const* d_in, const int* in_sizes, int n_in,
                              void* d_out, int out_size, void* d_ws,
                              size_t ws_size, hipStream_t stream) {
  const float* in = (const float*)d_in[0];
  float* out = (float*)d_out;
  const int nmat = in_sizes[0] / 4096;  // B*C = 8192 matrices of 64x64
  spd_logpool_exp_kernel<<<nmat, 256, 0, stream>>>(in, out);
}